// InterpolatingBSpline1d_88141318848861
// MI455X (gfx1250) — compile-verified
//
#include <hip/hip_runtime.h>

typedef __attribute__((ext_vector_type(2))) float v2f;
typedef __attribute__((ext_vector_type(8))) float v8f;

#define CCH 64  // channels (fixed by the reference problem)

// ---------------------------------------------------------------------------
// Kernel 1: spline coefficient solve.
// After natural-boundary elimination: c1 = d[0], cM = d[M-1], and interior
// unknowns c_2..c_{M-1} satisfy a constant tridiagonal (1/6, 2/3, 1/6)
// system with RHS data[1..M-2] (ends corrected by the known c1/cM).
// One thread per channel; the LU multipliers cp_i are channel-uniform and
// are shared through LDS for the backward pass.
// coefsT layout: row-major (M+2) x 64 (knot-major) for coalesced eval reads.
// ---------------------------------------------------------------------------
__global__ __launch_bounds__(64)
void bspline_solve_kernel(const float* __restrict__ data,
                          float* __restrict__ coefsT, int M) {
  const int ch = threadIdx.x;
  const float a = 1.0f / 6.0f;
  const float b = 2.0f / 3.0f;
  __shared__ float cp[1024];  // cp[i-2] for i = 2..M-1 (M <= 1026)

  const float* __restrict__ d = data + (size_t)ch * M;
  const float d0  = d[0];
  const float dM1 = d[M - 1];

  coefsT[1 * CCH + ch]         = d0;   // c1
  coefsT[(size_t)M * CCH + ch] = dM1;  // cM

  // Forward elimination (i = 2 .. M-1)
  float inv = 1.0f / b;
  float cpp = a * inv;
  float dpp = (d[1] - a * d0) * inv;
  if (ch == 0) cp[0] = cpp;
  coefsT[2 * CCH + ch] = dpp;

  #pragma unroll 4
  for (int i = 3; i <= M - 1; ++i) {
    float rhs = d[i - 1];
    if (i == M - 1) rhs -= a * dM1;
    float den = b - a * cpp;
    float dinv = 1.0f / den;
    cpp = a * dinv;
    dpp = (rhs - a * dpp) * dinv;
    if (ch == 0) cp[i - 2] = cpp;
    coefsT[(size_t)i * CCH + ch] = dpp;  // dp staged in-place
  }
  __syncthreads();  // wave 1 needs lane-0-written cp[]

  // Backward substitution (in place over coefsT)
  float x = dpp;          // x_{M-1} = dp_{M-1}  (already stored)
  const float xMm1 = x;   // c_{M-1}
  #pragma unroll 8
  for (int i = M - 2; i >= 2; --i) {
    float dpi = coefsT[(size_t)i * CCH + ch];
    x = dpi - cp[i - 2] * x;
    coefsT[(size_t)i * CCH + ch] = x;
  }
  // x now holds c_2; natural-boundary ghost rows:
  coefsT[0 * CCH + ch]               = 2.0f * d0 - x;      // c0  = 2c1 - c2
  coefsT[(size_t)(M + 1) * CCH + ch] = 2.0f * dM1 - xMm1;  // cM+1= 2cM - cM-1
}

// ---------------------------------------------------------------------------
// Kernel 2: evaluation. One wave32 per 16-sample tile.
// Weights for all 16 samples computed with ONE v_wmma_f32_16x16x4_f32:
//   A(16x4)  = t_powers  (lane m: [t^3, t^2] | lane 16+m: [t, 1])
//   B(4x16)  = BASIS^T   (cols 0..3 valid, rest zero)
//   D(16x16) -> D[s,k] = w[s,k] in the documented C/D layout.
// Each lane owns 2 channels (float2) so every tap read / sample store is a
// coalesced 256B row. Tap base addresses become wave-uniform via v_readlane.
// ---------------------------------------------------------------------------
__device__ __forceinline__ void eval_one_sample(
    const v8f& D, int i_lanes, int s, const float* __restrict__ coefsT,
    int c2, v2f* __restrict__ accOut) {
  const int srcHi = (s >> 3) << 4;
  const int j     = s & 7;

  const int   ib = __builtin_amdgcn_readlane(i_lanes, s);  // sample s base knot
  const float w0 = __builtin_bit_cast(
      float, __builtin_amdgcn_readlane(__builtin_bit_cast(int, D[j]), 0 + srcHi));
  const float w1 = __builtin_bit_cast(
      float, __builtin_amdgcn_readlane(__builtin_bit_cast(int, D[j]), 1 + srcHi));
  const float w2 = __builtin_bit_cast(
      float, __builtin_amdgcn_readlane(__builtin_bit_cast(int, D[j]), 2 + srcHi));
  const float w3 = __builtin_bit_cast(
      float, __builtin_amdgcn_readlane(__builtin_bit_cast(int, D[j]), 3 + srcHi));

  const float* base = coefsT + (size_t)ib * CCH + c2;
  const v2f p0 = *(const v2f*)(base + 0 * CCH);
  const v2f p1 = *(const v2f*)(base + 1 * CCH);
  const v2f p2 = *(const v2f*)(base + 2 * CCH);
  const v2f p3 = *(const v2f*)(base + 3 * CCH);

  v2f acc;
  acc.x = fmaf(w3, p3.x, fmaf(w2, p2.x, fmaf(w1, p1.x, w0 * p0.x)));
  acc.y = fmaf(w3, p3.y, fmaf(w2, p2.y, fmaf(w1, p1.y, w0 * p0.y)));
  *accOut = acc;
}

__global__ __launch_bounds__(256)
void bspline_eval_kernel(const float* __restrict__ u,
                         const float* __restrict__ coefsT,
                         float* __restrict__ out,
                         int B, int M, int ntiles) {
  const int lane = threadIdx.x & 31;
  const int tile = blockIdx.x * (blockDim.x >> 5) + (threadIdx.x >> 5);
  if (tile >= ntiles) return;  // wave-uniform exit -> EXEC all-ones at WMMA

  const int  sloc = lane & 15;
  const bool hiw  = lane >= 16;

  const int sidx   = tile * 16 + sloc;
  const int sclamp = (sidx < B) ? sidx : (B - 1);

  // stream u ahead (gfx1250 global_prefetch_b8; speculative, OOB-safe)
  __builtin_prefetch(u + sclamp + 4096, 0, 0);

  const float uu = u[sclamp];
  float un = uu * (float)(M - 1);
  float fi = floorf(un);
  int   i  = (int)fi;
  float t  = un - fi;
  if (un < 0.0f)            { i = 0;     t = un; }
  if (un >= (float)(M - 2)) { i = M - 2; t = un - (float)(M - 2); }

  const float t2 = t * t;
  const float t3 = t2 * t;

  // A-matrix 16x4 f32 layout: VGPR0 = K0 | K2, VGPR1 = K1 | K3
  v2f Am;
  Am.x = hiw ? t    : t3;
  Am.y = hiw ? 1.0f : t2;

  // B-matrix 4x16 f32 layout: VGPR0 = K0 (lanes 0-15) | K2 (16-31),
  //                           VGPR1 = K1 | K3 ;  B[j,n] = BASIS[n][j]
  float b0 = 0.0f, b1 = 0.0f;
  const int n = sloc;
  if      (n == 0) { b0 = hiw ? -3.0f :  -1.0f; b1 = hiw ? 1.0f :  3.0f; }
  else if (n == 1) { b0 = hiw ?  0.0f :   3.0f; b1 = hiw ? 4.0f : -6.0f; }
  else if (n == 2) { b0 = hiw ?  3.0f :  -3.0f; b1 = hiw ? 1.0f :  3.0f; }
  else if (n == 3) { b0 = hiw ?  0.0f :   1.0f; b1 = hiw ? 0.0f :  0.0f; }
  const float s6 = 1.0f / 6.0f;
  v2f Bm;
  Bm.x = b0 * s6;
  Bm.y = b1 * s6;

  v8f Cz = {0.f, 0.f, 0.f, 0.f, 0.f, 0.f, 0.f, 0.f};
  // D = A x B + 0 : weights for 16 samples in one matrix op
  v8f D = __builtin_amdgcn_wmma_f32_16x16x4_f32(
      /*neg_a=*/false, Am, /*neg_b=*/false, Bm,
      /*c_mod=*/(short)0, Cz, /*reuse_a=*/false, /*reuse_b=*/false);
  // D layout: lane L, VGPR j -> element (M = j + 8*(L>=16), N = L%16)
  //           => w[sample s, tap k] lives in VGPR (s&7), lane (k + 16*(s>>3))

  const int c2      = lane * 2;  // this lane's channel pair
  float*    outTile = out + (size_t)tile * 16 * CCH;

  if (tile * 16 + 16 <= B) {
    // Full tile (the only path when B % 16 == 0): branch-free, NT streaming
    // stores — output is write-once, never re-read.
    #pragma unroll
    for (int s = 0; s < 16; ++s) {
      v2f acc;
      eval_one_sample(D, i, s, coefsT, c2, &acc);
      __builtin_nontemporal_store(acc, (v2f*)(outTile + s * CCH + c2));
    }
  } else {
    // Single possible tail tile: per-sample guard is fine here.
    #pragma unroll
    for (int s = 0; s < 16; ++s) {
      v2f acc;
      eval_one_sample(D, i, s, coefsT, c2, &acc);
      if (tile * 16 + s < B) {
        __builtin_nontemporal_store(acc, (v2f*)(outTile + s * CCH + c2));
      }
    }
  }
}

// ---------------------------------------------------------------------------
extern "C" void kernel_launch(void* const* d_in, const int* in_sizes, int n_in,
                              void* d_out, int out_size, void* d_ws, size_t ws_size,
                              hipStream_t stream) {
  const float* data = (const float*)d_in[0];   // (C=64, M) f32
  const float* u    = (const float*)d_in[1];   // (B, 1)   f32
  float*       out  = (float*)d_out;           // (B, 64)  f32

  const int C = CCH;
  const int M = in_sizes[0] / C;               // 1024
  const int B = in_sizes[1];                   // 500000

  // workspace: coefsT, (M+2) x 64 f32 == ~263 KB
  float* coefsT = (float*)d_ws;

  bspline_solve_kernel<<<1, 64, 0, stream>>>(data, coefsT, M);

  const int ntiles        = (B + 15) / 16;
  const int wavesPerBlock = 8;                 // 256 threads
  const int blocks        = (ntiles + wavesPerBlock - 1) / wavesPerBlock;
  bspline_eval_kernel<<<blocks, 256, 0, stream>>>(u, coefsT, out, B, M, ntiles);
}